// Generator3DLUT_identity_231928234068
// MI455X (gfx1250) — compile-verified
//
#include <hip/hip_runtime.h>
#include <stdint.h>

// ---------------------------------------------------------------------------
// Trilinear 3D LUT apply (33^3 LUT, 8x3x1024x1024 fp32 image).
// Memory-bound: ~193 MiB mandatory HBM traffic -> ~8.3us floor @ 23.3 TB/s.
// Strategy: pack LUT as half4 (8B/entry, 287.5KB) -> fits 320KB LDS/WGP;
// persistent blocks fill LDS once via CDNA5 async global->LDS loads
// (global_load_async_to_lds_b128 + s_wait_asynccnt), then stream pixels with
// b128 nontemporal loads/stores; the 8 divergent corner gathers per pixel
// come from LDS (compiler pairs them into ds_load_2addr_b64).
// ---------------------------------------------------------------------------

#define DIM      33
#define DIM2     (DIM * DIM)            // 1089
#define NLUT     (DIM * DIM * DIM)      // 35937
#define NLUT_PAD 35938                  // even -> LDS image is a multiple of 16B
#define PLANE    (1024 * 1024)
#define NBATCH   8
#define CHUNKS   ((NBATCH * PLANE) / 4) // 4 pixels per thread-iteration
#define BLOCK    1024
#define GRID     512
#define ITERS    (CHUNKS / (GRID * BLOCK))

static_assert(CHUNKS == ITERS * GRID * BLOCK, "exact tiling, no tail");

typedef float    v4f __attribute__((ext_vector_type(4)));
typedef uint32_t v2u __attribute__((ext_vector_type(2)));
typedef uint32_t v4u __attribute__((ext_vector_type(4)));
typedef _Float16 v2h __attribute__((ext_vector_type(2)));
typedef int      v4i __attribute__((ext_vector_type(4)));

// Exact pointer types for the async-to-LDS builtin (per hipcc diagnostic:
// first arg is "int __vector(4) __device__ *", i.e. v4i in global AS).
typedef __attribute__((address_space(1))) v4i gv4i;
typedef __attribute__((address_space(3))) v4i lv4i;

// ---- main kernel (placed first so the disasm snippet shows it) ------------
__global__ __launch_bounds__(BLOCK) void lut3d_lds_kernel(
    const float* __restrict__ x, const v2u* __restrict__ packed,
    float* __restrict__ out) {
  __shared__ v2u slut[NLUT_PAD];  // 287,504 bytes of the 320KB/WGP LDS

  // One-time LDS fill per (persistent) workgroup.
  {
    constexpr int NV = (NLUT_PAD * 8) / 16;  // 17969 x 16B transfers
#if __has_builtin(__builtin_amdgcn_global_load_async_to_lds_b128)
    for (int i = threadIdx.x; i < NV; i += BLOCK) {
      __builtin_amdgcn_global_load_async_to_lds_b128(
          (gv4i*)((const char*)packed + 16 * i),
          (lv4i*)((char*)slut + 16 * i), 0, 0);
    }
#if __has_builtin(__builtin_amdgcn_s_wait_asynccnt)
    __builtin_amdgcn_s_wait_asynccnt(0);
#else
    asm volatile("s_wait_asynccnt 0" ::: "memory");
#endif
#else
    const v4u* __restrict__ src = (const v4u*)packed;
    v4u* dst = (v4u*)slut;
    for (int i = threadIdx.x; i < NV; i += BLOCK) dst[i] = src[i];
#endif
    __syncthreads();
  }

  // rs = r / binsize in the reference; 1/binsize as a constant. Trilinear
  // interpolation is continuous across bin boundaries, so the <=1ulp
  // difference vs a true divide is numerically irrelevant.
  const float invbin = 1.0f / (float)(1.000001 / 32.0);
  const int tid0 = blockIdx.x * BLOCK + threadIdx.x;

#pragma unroll
  for (int it = 0; it < ITERS; ++it) {
    const int chunk = tid0 + it * (GRID * BLOCK);
    const int n = chunk >> 18;                   // PLANE/4 == 2^18
    const int q = (chunk & ((PLANE / 4) - 1)) << 2;
    const float* __restrict__ pin = x + (size_t)n * (3 * PLANE) + q;

    v4f r4 = __builtin_nontemporal_load((const v4f*)pin);
    v4f g4 = __builtin_nontemporal_load((const v4f*)(pin + PLANE));
    v4f b4 = __builtin_nontemporal_load((const v4f*)(pin + 2 * PLANE));
    v4f o_r, o_g, o_b;

#pragma unroll
    for (int k = 0; k < 4; ++k) {
      float rs = r4[k] * invbin;
      float gs = g4[k] * invbin;
      float bs = b4[k] * invbin;
      int ri = (int)rs; ri = ri > 31 ? 31 : ri;   // x in [0,1): ri<=31 anyway
      int gi = (int)gs; gi = gi > 31 ? 31 : gi;
      int bi = (int)bs; bi = bi > 31 ? 31 : bi;
      float rd = rs - (float)ri;
      float gd = gs - (float)gi;
      float bd = bs - (float)bi;
      int base = (bi * DIM + gi) * DIM + ri;
      float wr0 = 1.0f - rd, wg0 = 1.0f - gd, wb0 = 1.0f - bd;
      float w00 = wb0 * wg0, w01 = wb0 * gd, w10 = bd * wg0, w11 = bd * gd;
      float accr = 0.0f, accg = 0.0f, accb = 0.0f;
      auto corner = [&](int idx, float w) {
        v2u e = slut[idx];                        // ds_load (2addr-paired)
        union { uint32_t u; v2h h; } lo, hi;
        lo.u = e.x;
        hi.u = e.y;
        accr += w * (float)lo.h.x;
        accg += w * (float)lo.h.y;
        accb += w * (float)hi.h.x;
      };
      // same accumulation order as the reference: (db,dg,dr) loops
      corner(base,               w00 * wr0);
      corner(base + 1,           w00 * rd);
      corner(base + DIM,         w01 * wr0);
      corner(base + DIM + 1,     w01 * rd);
      corner(base + DIM2,        w10 * wr0);
      corner(base + DIM2 + 1,    w10 * rd);
      corner(base + DIM2 + DIM,  w11 * wr0);
      corner(base + DIM2 + DIM + 1, w11 * rd);
      o_r[k] = accr;
      o_g[k] = accg;
      o_b[k] = accb;
    }

    float* __restrict__ pout = out + (size_t)n * (3 * PLANE) + q;
    __builtin_nontemporal_store(o_r, (v4f*)pout);
    __builtin_nontemporal_store(o_g, (v4f*)(pout + PLANE));
    __builtin_nontemporal_store(o_b, (v4f*)(pout + 2 * PLANE));
  }
}

// ---- kernel 1: interleave LUT (3 fp32 planes) -> half4 entries in d_ws ----
__global__ void pack_lut_kernel(const float* __restrict__ lut,
                                v2u* __restrict__ packed) {
  int i = blockIdx.x * blockDim.x + threadIdx.x;
  if (i >= NLUT_PAD) return;
  v2u e = {0u, 0u};
  if (i < NLUT) {
    union { v2h h; uint32_t u; } lo, hi;
    lo.h = (v2h){(_Float16)lut[i], (_Float16)lut[i + NLUT]};
    hi.h = (v2h){(_Float16)lut[i + 2 * NLUT], (_Float16)0.0f};
    e.x = lo.u;
    e.y = hi.u;
  }
  packed[i] = e;
}

// ---- fallback: direct fp32 global gathers (if workspace is too small) -----
__global__ __launch_bounds__(BLOCK) void lut3d_global_kernel(
    const float* __restrict__ x, const float* __restrict__ lut,
    float* __restrict__ out) {
  const float invbin = 1.0f / (float)(1.000001 / 32.0);
  const int tid0 = blockIdx.x * BLOCK + threadIdx.x;

  for (int it = 0; it < ITERS; ++it) {
    const int chunk = tid0 + it * (GRID * BLOCK);
    const int n = chunk >> 18;
    const int q = (chunk & ((PLANE / 4) - 1)) << 2;
    const float* __restrict__ pin = x + (size_t)n * (3 * PLANE) + q;

    v4f r4 = __builtin_nontemporal_load((const v4f*)pin);
    v4f g4 = __builtin_nontemporal_load((const v4f*)(pin + PLANE));
    v4f b4 = __builtin_nontemporal_load((const v4f*)(pin + 2 * PLANE));
    v4f o_r, o_g, o_b;

#pragma unroll
    for (int k = 0; k < 4; ++k) {
      float rs = r4[k] * invbin;
      float gs = g4[k] * invbin;
      float bs = b4[k] * invbin;
      int ri = (int)rs; ri = ri > 31 ? 31 : ri;
      int gi = (int)gs; gi = gi > 31 ? 31 : gi;
      int bi = (int)bs; bi = bi > 31 ? 31 : bi;
      float rd = rs - (float)ri;
      float gd = gs - (float)gi;
      float bd = bs - (float)bi;
      int base = (bi * DIM + gi) * DIM + ri;
      float wr0 = 1.0f - rd, wg0 = 1.0f - gd, wb0 = 1.0f - bd;
      float w00 = wb0 * wg0, w01 = wb0 * gd, w10 = bd * wg0, w11 = bd * gd;
      float accr = 0.0f, accg = 0.0f, accb = 0.0f;
      auto corner = [&](int idx, float w) {
        accr += w * lut[idx];
        accg += w * lut[idx + NLUT];
        accb += w * lut[idx + 2 * NLUT];
      };
      corner(base,               w00 * wr0);
      corner(base + 1,           w00 * rd);
      corner(base + DIM,         w01 * wr0);
      corner(base + DIM + 1,     w01 * rd);
      corner(base + DIM2,        w10 * wr0);
      corner(base + DIM2 + 1,    w10 * rd);
      corner(base + DIM2 + DIM,  w11 * wr0);
      corner(base + DIM2 + DIM + 1, w11 * rd);
      o_r[k] = accr;
      o_g[k] = accg;
      o_b[k] = accb;
    }

    float* __restrict__ pout = out + (size_t)n * (3 * PLANE) + q;
    __builtin_nontemporal_store(o_r, (v4f*)pout);
    __builtin_nontemporal_store(o_g, (v4f*)(pout + PLANE));
    __builtin_nontemporal_store(o_b, (v4f*)(pout + 2 * PLANE));
  }
}

extern "C" void kernel_launch(void* const* d_in, const int* in_sizes, int n_in,
                              void* d_out, int out_size, void* d_ws,
                              size_t ws_size, hipStream_t stream) {
  (void)in_sizes; (void)n_in; (void)out_size;
  const float* lut = (const float*)d_in[0];  // (3,33,33,33) fp32
  const float* x   = (const float*)d_in[1];  // (8,3,1024,1024) fp32
  float* out       = (float*)d_out;          // (8,3,1024,1024) fp32

  const size_t need = (size_t)NLUT_PAD * 8;  // packed half4 LUT bytes
  if (ws_size >= need) {
    v2u* packed = (v2u*)d_ws;
    pack_lut_kernel<<<(NLUT_PAD + 255) / 256, 256, 0, stream>>>(lut, packed);
    lut3d_lds_kernel<<<GRID, BLOCK, 0, stream>>>(x, packed, out);
  } else {
    lut3d_global_kernel<<<GRID, BLOCK, 0, stream>>>(x, lut, out);
  }
}